// ITEMBRANCH_58514634441035
// MI455X (gfx1250) — compile-verified
//
#include <hip/hip_runtime.h>
#include <hip/hip_bf16.h>

#define TOTAL_K 800000
#define N_HEAD  100000
#define HIDDEN  128
#define DENOM   (1.0f / (12800000.0f))   // N_HEAD * HIDDEN

typedef float v2f __attribute__((ext_vector_type(2)));
typedef float v8f __attribute__((ext_vector_type(8)));

// ---------------------------------------------------------------- zero scratch
__global__ void itembranch_zero(float* p, int n) {
    int i = blockIdx.x * blockDim.x + threadIdx.x;
    if (i < n) p[i] = 0.0f;
}

// ---------------------------------------------------------------- counts
__global__ void itembranch_counts(const int* __restrict__ segids,
                                  float* __restrict__ counts) {
    int i = blockIdx.x * blockDim.x + threadIdx.x;
    if (i < TOTAL_K) atomicAdd(&counts[segids[i]], 1.0f);
}

// ---------------------------------------------------------------- adapter GEMM
// x = s + s @ W_U^T + b, scattered into per-segment sums via f32 atomics.
// Block = 128 threads = 4 waves; each wave owns a 16-row M-tile (64 rows/block).
// W_U (64KB) staged into LDS once per block; B fragments via ds_load_b64.
__global__ void __launch_bounds__(128)
itembranch_adapter(const float* __restrict__ subseq,
                   const float* __restrict__ WU,
                   const float* __restrict__ bU,
                   const int*   __restrict__ segids,
                   float*       __restrict__ sums) {
    __shared__ float sW[HIDDEN * HIDDEN];   // 64 KB
    const int tid  = threadIdx.x;
    const int wave = tid >> 5;
    const int lane = tid & 31;
    const int lh   = lane >> 4;   // half-wave select
    const int ll   = lane & 15;

    // stage W_U into LDS with float4
    {
        const float4* g = (const float4*)WU;
        float4*       l = (float4*)sW;
        for (int i = tid; i < (HIDDEN * HIDDEN) / 4; i += 128) l[i] = g[i];
    }
    __syncthreads();

    const int row0 = blockIdx.x * 64 + wave * 16;   // TOTAL_K % 64 == 0

    v8f acc[8];
    #pragma unroll
    for (int nt = 0; nt < 8; ++nt) acc[nt] = (v8f)0.0f;

    const int arow = row0 + ll;                     // A: M striped over lanes
    const float* aBase = subseq + (size_t)arow * HIDDEN;

    for (int kt = 0; kt < 32; ++kt) {
        const int k = kt * 4 + 2 * lh;              // this lane's K pair
        v2f a = *(const v2f*)(aBase + k);           // global_load_b64
        #pragma unroll
        for (int nt = 0; nt < 8; ++nt) {
            const int n = nt * 16 + ll;
            // B[k][n] = W_U[n][k]  (s @ W^T)
            v2f b = *(const v2f*)(&sW[n * HIDDEN + k]);   // ds_load_b64
            acc[nt] = __builtin_amdgcn_wmma_f32_16x16x4_f32(
                false, a, false, b, (short)0, acc[nt], false, false);
        }
    }

    // epilogue: residual + bias, scatter-add into segment sums
    int seg[8];
    #pragma unroll
    for (int r = 0; r < 8; ++r) seg[r] = segids[row0 + r + 8 * lh];

    #pragma unroll
    for (int nt = 0; nt < 8; ++nt) {
        const int col  = nt * 16 + ll;
        const float bb = bU[col];
        #pragma unroll
        for (int r = 0; r < 8; ++r) {
            const int grow = row0 + r + 8 * lh;     // C/D: M = r + 8*(lane>=16)
            float val = acc[nt][r] + subseq[(size_t)grow * HIDDEN + col] + bb;
            atomicAdd(&sums[(size_t)seg[r] * HIDDEN + col], val);
        }
    }
}

// ---------------------------------------------------------------- head GEMM + loss
// pred = (sums/cnt) @ W_I^T + b; loss partial = w_i * (pred - tgt)^2.
__global__ void __launch_bounds__(128)
itembranch_head(const float* __restrict__ sums,
                const float* __restrict__ counts,
                const float* __restrict__ WI,
                const float* __restrict__ bI,
                const float* __restrict__ item_emb,
                const float* __restrict__ w_i,
                const int*   __restrict__ target_idx,
                float*       __restrict__ lossAcc) {
    __shared__ float sW[HIDDEN * HIDDEN];   // 64 KB
    const int tid  = threadIdx.x;
    const int wave = tid >> 5;
    const int lane = tid & 31;
    const int lh   = lane >> 4;
    const int ll   = lane & 15;

    {
        const float4* g = (const float4*)WI;
        float4*       l = (float4*)sW;
        for (int i = tid; i < (HIDDEN * HIDDEN) / 4; i += 128) l[i] = g[i];
    }
    __syncthreads();

    const int row0 = blockIdx.x * 64 + wave * 16;

    // A-side head index per lane (clamp tail so EXEC stays all-1s for WMMA)
    const int hA = min(row0 + ll, N_HEAD - 1);
    const float inv = 1.0f / fmaxf(counts[hA], 1.0f);
    const float* aBase = sums + (size_t)hA * HIDDEN;

    v8f acc[8];
    #pragma unroll
    for (int nt = 0; nt < 8; ++nt) acc[nt] = (v8f)0.0f;

    for (int kt = 0; kt < 32; ++kt) {
        const int k = kt * 4 + 2 * lh;
        v2f a = *(const v2f*)(aBase + k);
        a.x *= inv; a.y *= inv;                     // mean pooling fused in
        #pragma unroll
        for (int nt = 0; nt < 8; ++nt) {
            const int n = nt * 16 + ll;
            v2f b = *(const v2f*)(&sW[n * HIDDEN + k]);
            acc[nt] = __builtin_amdgcn_wmma_f32_16x16x4_f32(
                false, a, false, b, (short)0, acc[nt], false, false);
        }
    }

    float partial = 0.0f;
    #pragma unroll
    for (int nt = 0; nt < 8; ++nt) {
        const int col  = nt * 16 + ll;
        const float bb = bI[col];
        #pragma unroll
        for (int r = 0; r < 8; ++r) {
            const int head = row0 + r + 8 * lh;
            if (head < N_HEAD) {
                const int tix = target_idx[head];
                const float tgt = item_emb[(size_t)tix * HIDDEN + col];
                const float d = acc[nt][r] + bb - tgt;
                partial += w_i[head] * d * d;
            }
        }
    }
    atomicAdd(lossAcc, partial);
}

// ---------------------------------------------------------------- finalize
__global__ void itembranch_finalize(const float* __restrict__ acc,
                                    float* __restrict__ out) {
    if (threadIdx.x == 0 && blockIdx.x == 0) out[0] = acc[0] * DENOM;
}

// ---------------------------------------------------------------- launcher
extern "C" void kernel_launch(void* const* d_in, const int* in_sizes, int n_in,
                              void* d_out, int out_size, void* d_ws, size_t ws_size,
                              hipStream_t stream) {
    const float* subseq     = (const float*)d_in[0];
    const float* W_U_w      = (const float*)d_in[1];
    const float* W_U_b      = (const float*)d_in[2];
    const float* W_I_w      = (const float*)d_in[3];
    const float* W_I_b      = (const float*)d_in[4];
    const float* item_emb   = (const float*)d_in[5];
    const float* w_i        = (const float*)d_in[6];
    const int*   segids     = (const int*)d_in[7];
    const int*   target_idx = (const int*)d_in[8];
    float* out = (float*)d_out;

    // ws layout: sums [N_HEAD*HIDDEN] | counts [N_HEAD] | lossAcc [1]
    float* sums    = (float*)d_ws;
    float* counts  = sums + (size_t)N_HEAD * HIDDEN;
    float* lossAcc = counts + N_HEAD;

    const int nz = N_HEAD * HIDDEN + N_HEAD + 1;
    itembranch_zero<<<(nz + 255) / 256, 256, 0, stream>>>((float*)d_ws, nz);

    itembranch_adapter<<<TOTAL_K / 64, 128, 0, stream>>>(
        subseq, W_U_w, W_U_b, segids, sums);

    itembranch_counts<<<(TOTAL_K + 255) / 256, 256, 0, stream>>>(segids, counts);

    itembranch_head<<<(N_HEAD + 63) / 64, 128, 0, stream>>>(
        sums, counts, W_I_w, W_I_b, item_emb, w_i, target_idx, lossAcc);

    itembranch_finalize<<<1, 1, 0, stream>>>(lossAcc, out);
}